// JKNet_56307021250669
// MI455X (gfx1250) — compile-verified
//
#include <hip/hip_runtime.h>

// ---------------------------------------------------------------------------
// JKNet GCN forward for MI455X (gfx1250, wave32).
// GEMMs via V_WMMA_F32_16X16X4_F32 (fp32 matrix pipe, matches fp32 reference).
// Aggregation via fp32 global atomics (L2-resident working set).
// ---------------------------------------------------------------------------

typedef __attribute__((ext_vector_type(2))) float v2f;
typedef __attribute__((ext_vector_type(8))) float v8f;

#define N_NODES_F96 96
#define HID 96
#define D_OUT 64

// ---------------------------- degree / dinv --------------------------------

__global__ void deg_init_kernel(float* __restrict__ deg, int n) {
    int i = blockIdx.x * blockDim.x + threadIdx.x;
    if (i < n) deg[i] = 1.0f;  // self-loop contributes 1 to every node's degree
}

__global__ void deg_scatter_kernel(const int* __restrict__ dst,
                                   float* __restrict__ deg, int E) {
    int i = blockIdx.x * blockDim.x + threadIdx.x;
    if (i < E) atomicAdd(&deg[dst[i]], 1.0f);
}

__global__ void deg_rsqrt_kernel(float* __restrict__ deg, int n) {
    int i = blockIdx.x * blockDim.x + threadIdx.x;
    if (i < n) {
        float d = deg[i];
        deg[i] = (d > 0.0f) ? rsqrtf(d) : 0.0f;  // deg >= 1 always, but match ref
    }
}

// ------------------------------ WMMA GEMM ----------------------------------
// C[M,N] = A[M,K] @ W[K,N], row-major, fp32. One wave computes one 16x16 tile.
// A 16x4 layout : lane(0-15)=row M, VGPR0/1 hold K = 2*half + {0,1}.
// B 4x16 layout : lane(0-15)=col N, VGPR0/1 hold K = 2*half + {0,1}.
// C 16x16 layout: lane(0-15)=col N, VGPR r holds row = 8*half + r.
__global__ void gemm_wmma_kernel(const float* __restrict__ A,
                                 const float* __restrict__ W,
                                 float* __restrict__ C,
                                 int M, int K, int N) {
    int wave = (int)((blockIdx.x * blockDim.x + threadIdx.x) >> 5);
    int ntn  = N >> 4;
    int mt   = wave / ntn;
    int nt   = wave - mt * ntn;
    if (mt * 16 >= M) return;

    int lane = threadIdx.x & 31;
    int half = lane >> 4;      // 0: K even pair base, rows 0-7 of C
    int l15  = lane & 15;

    const float* Arow = A + (size_t)(mt * 16 + l15) * K + 2 * half;
    const float* Bp   = W + (size_t)(2 * half) * N + nt * 16 + l15;

    v8f acc = {};
    for (int k = 0; k < K; k += 4) {
        v2f a, b;
        a.x = Arow[k];
        a.y = Arow[k + 1];
        b.x = Bp[(size_t)k * N];
        b.y = Bp[(size_t)(k + 1) * N];
        acc = __builtin_amdgcn_wmma_f32_16x16x4_f32(
            /*neg_a=*/false, a, /*neg_b=*/false, b,
            /*c_mod=*/(short)0, acc, /*reuse_a=*/false, /*reuse_b=*/false);
    }

    float* Crow = C + (size_t)(mt * 16 + half * 8) * N + nt * 16 + l15;
#pragma unroll
    for (int r = 0; r < 8; ++r) Crow[(size_t)r * N] = acc[r];
}

// JK layer: C[M,64] = sum_{layer=0..4} H_layer[M,96] @ Wout[layer*96:(l+1)*96, 64]
// H layers are contiguous blocks of M*96 starting at Hbase (the d_out h slots),
// so the 480-wide concatenated matrix is never materialized.
__global__ void gemm_cat_wmma_kernel(const float* __restrict__ Hbase,
                                     const float* __restrict__ Wout,
                                     float* __restrict__ C, int M) {
    const int K1 = HID;     // 96 per block
    const int N  = D_OUT;   // 64
    int wave = (int)((blockIdx.x * blockDim.x + threadIdx.x) >> 5);
    int ntn  = N >> 4;      // 4
    int mt   = wave / ntn;
    int nt   = wave - mt * ntn;
    if (mt * 16 >= M) return;

    int lane = threadIdx.x & 31;
    int half = lane >> 4;
    int l15  = lane & 15;

    v8f acc = {};
    for (int layer = 0; layer < 5; ++layer) {
        const float* Arow = Hbase + (size_t)layer * M * K1
                                  + (size_t)(mt * 16 + l15) * K1 + 2 * half;
        const float* Bp   = Wout + (size_t)(layer * K1 + 2 * half) * N
                                  + nt * 16 + l15;
        for (int k = 0; k < K1; k += 4) {
            v2f a, b;
            a.x = Arow[k];
            a.y = Arow[k + 1];
            b.x = Bp[(size_t)k * N];
            b.y = Bp[(size_t)(k + 1) * N];
            acc = __builtin_amdgcn_wmma_f32_16x16x4_f32(
                false, a, false, b, (short)0, acc, false, false);
        }
    }

    float* Crow = C + (size_t)(mt * 16 + half * 8) * N + nt * 16 + l15;
#pragma unroll
    for (int r = 0; r < 8; ++r) Crow[(size_t)r * N] = acc[r];
}

// --------------------------- aggregation pass ------------------------------

// agg[v,f] = hlin[v,f] * dinv[v]^2   (self-loop message, also initializes agg)
__global__ void agg_init_kernel(const float* __restrict__ hlin,
                                const float* __restrict__ dinv,
                                float* __restrict__ agg, int F) {
    int v = blockIdx.x;
    int f = threadIdx.x;
    float w = dinv[v];
    w *= w;
    agg[(size_t)v * F + f] = hlin[(size_t)v * F + f] * w;
}

// One wave per edge: agg[dst] += hlin[src] * dinv[src]*dinv[dst]
__global__ void edge_scatter_kernel(const int* __restrict__ src,
                                    const int* __restrict__ dst,
                                    const float* __restrict__ dinv,
                                    const float* __restrict__ hlin,
                                    float* __restrict__ agg,
                                    int E, int F) {
    int e = (int)((blockIdx.x * blockDim.x + threadIdx.x) >> 5);
    if (e >= E) return;
    int lane = threadIdx.x & 31;
    int s = src[e];
    int d = dst[e];
    float w = dinv[s] * dinv[d];
    const float* hs = hlin + (size_t)s * F;
    float*       ad = agg  + (size_t)d * F;
    for (int f = lane; f < F; f += 32)
        atomicAdd(&ad[f], hs[f] * w);
}

// dest[v,f] = (relu ? max(agg+b, 0) : agg+b)
__global__ void finalize_kernel(const float* __restrict__ agg,
                                const float* __restrict__ bias,
                                float* __restrict__ dest, int F, int do_relu) {
    int v = blockIdx.x;
    int f = threadIdx.x;
    float val = agg[(size_t)v * F + f] + bias[f];
    if (do_relu) val = fmaxf(val, 0.0f);
    dest[(size_t)v * F + f] = val;
}

// ------------------------------- launcher ----------------------------------

extern "C" void kernel_launch(void* const* d_in, const int* in_sizes, int n_in,
                              void* d_out, int out_size, void* d_ws, size_t ws_size,
                              hipStream_t stream) {
    const float* x    = (const float*)d_in[0];   // [M, 96]
    const int*   ei   = (const int*)d_in[1];     // [2, E]
    // d_in[2] percent, d_in[3] ricci_curvature: unused on eval path
    const float* W1   = (const float*)d_in[4];   // [96, 96]
    const float* b1   = (const float*)d_in[5];   // [96]
    const float* Ws   = (const float*)d_in[6];   // [4, 96, 96]
    const float* bs   = (const float*)d_in[7];   // [4, 96]
    const float* Wout = (const float*)d_in[8];   // [480, 64]
    const float* bout = (const float*)d_in[9];   // [64]
    float* out = (float*)d_out;

    const int M = in_sizes[0] / N_NODES_F96;     // 50000 (multiple of 16)
    const int E = in_sizes[1] / 2;               // 800000
    const int* src = ei;
    const int* dst = ei + E;

    // workspace layout (floats): deg/dinv | hlin | agg
    float* ws   = (float*)d_ws;
    float* deg  = ws;                                        // M (becomes dinv)
    float* hlin = ws + (((size_t)M + 255) & ~(size_t)255);   // M*96
    float* agg  = hlin + (size_t)M * HID;                    // M*96

    // h1..h5 live directly in their d_out slots, after out[M,64]
    float* hbase = out + (size_t)M * D_OUT;

    // ---- degree / dinv ----
    deg_init_kernel<<<(M + 255) / 256, 256, 0, stream>>>(deg, M);
    deg_scatter_kernel<<<(E + 255) / 256, 256, 0, stream>>>(dst, deg, E);
    deg_rsqrt_kernel<<<(M + 255) / 256, 256, 0, stream>>>(deg, M);

    const int tiles96 = (M / 16) * (HID / 16);    // 18750 wave-tiles
    const int gemm_grid96 = (tiles96 + 3) / 4;    // 4 waves (128 thr) / block
    const int scat_grid = (E + 7) / 8;            // 8 edges (waves) / block

    // ---- conv for hidden layers (F = 96) ----
    auto conv96 = [&](const float* Ain, const float* Wp, const float* bp,
                      float* dest) {
        gemm_wmma_kernel<<<gemm_grid96, 128, 0, stream>>>(Ain, Wp, hlin,
                                                          M, HID, HID);
        agg_init_kernel<<<M, HID, 0, stream>>>(hlin, deg, agg, HID);
        edge_scatter_kernel<<<scat_grid, 256, 0, stream>>>(src, dst, deg,
                                                           hlin, agg, E, HID);
        finalize_kernel<<<M, HID, 0, stream>>>(agg, bp, dest, HID, 1);
    };

    // layer 1: x @ W1
    conv96(x, W1, b1, hbase);
    // hidden layers 2..5: h_i @ Ws[i]
    for (int i = 0; i < 4; ++i) {
        conv96(hbase + (size_t)i * M * HID,
               Ws + (size_t)i * HID * HID,
               bs + (size_t)i * HID,
               hbase + (size_t)(i + 1) * M * HID);
    }

    // ---- JK output conv (F = 64, no relu) ----
    const int tiles64 = (M / 16) * (D_OUT / 16);  // 12500
    gemm_cat_wmma_kernel<<<(tiles64 + 3) / 4, 128, 0, stream>>>(hbase, Wout,
                                                                hlin, M);
    agg_init_kernel<<<M, D_OUT, 0, stream>>>(hlin, deg, agg, D_OUT);
    edge_scatter_kernel<<<scat_grid, 256, 0, stream>>>(src, dst, deg,
                                                       hlin, agg, E, D_OUT);
    finalize_kernel<<<M, D_OUT, 0, stream>>>(agg, bout, out, D_OUT, 0);
}